// Criterion_46986942218249
// MI455X (gfx1250) — compile-verified
//
#include <hip/hip_runtime.h>
#include <hip/hip_bf16.h>

typedef __attribute__((ext_vector_type(2))) float    v2f;
typedef __attribute__((ext_vector_type(8))) float    v8f;
typedef __attribute__((ext_vector_type(4))) unsigned u32x4;
typedef __attribute__((ext_vector_type(8))) int      i32x8;
typedef __attribute__((ext_vector_type(4))) int      i32x4;

#define EPS_C      0.004f
#define WEIGHT_C   5000.0f
#define NORM_EPS_C 1e-12f

// ---------------------------------------------------------------------------
// K0: per (batch, face) precompute:
//   packedA[b][f]   = (-2cx, -2cy, -2cz, |c|^2)   from obstacle_pos centroids
//   faceNextC[b][f] = centroid of obstacle_next verts (xyz, 0)
//   faceNorm[b][f]  = normalized face normal of obstacle_next (xyz, 0)
// ---------------------------------------------------------------------------
__global__ void __launch_bounds__(256)
precompute_faces(const float* __restrict__ obsCurr,
                 const float* __restrict__ obsNext,
                 const int*   __restrict__ faces,
                 float* __restrict__ packedA,
                 float* __restrict__ faceNextC,
                 float* __restrict__ faceNorm,
                 int B, int No, int F)
{
    int idx = blockIdx.x * blockDim.x + threadIdx.x;
    if (idx >= B * F) return;
    int b = idx / F;
    int f = idx - b * F;

    int i0 = faces[3 * f + 0];
    int i1 = faces[3 * f + 1];
    int i2 = faces[3 * f + 2];

    const float third = 1.0f / 3.0f;

    {
        const float* oc = obsCurr + (size_t)b * No * 3;
        float cx = (oc[3*i0+0] + oc[3*i1+0] + oc[3*i2+0]) * third;
        float cy = (oc[3*i0+1] + oc[3*i1+1] + oc[3*i2+1]) * third;
        float cz = (oc[3*i0+2] + oc[3*i1+2] + oc[3*i2+2]) * third;
        float* pa = packedA + (size_t)idx * 4;
        pa[0] = -2.0f * cx;
        pa[1] = -2.0f * cy;
        pa[2] = -2.0f * cz;
        pa[3] = cx * cx + cy * cy + cz * cz;
    }

    {
        const float* on = obsNext + (size_t)b * No * 3;
        float v0x = on[3*i0+0], v0y = on[3*i0+1], v0z = on[3*i0+2];
        float v1x = on[3*i1+0], v1y = on[3*i1+1], v1z = on[3*i1+2];
        float v2x = on[3*i2+0], v2y = on[3*i2+1], v2z = on[3*i2+2];

        float* fc = faceNextC + (size_t)idx * 4;
        fc[0] = (v0x + v1x + v2x) * third;
        fc[1] = (v0y + v1y + v2y) * third;
        fc[2] = (v0z + v1z + v2z) * third;
        fc[3] = 0.0f;

        float e1x = v1x - v0x, e1y = v1y - v0y, e1z = v1z - v0z;
        float e2x = v2x - v0x, e2y = v2y - v0y, e2z = v2z - v0z;
        float nx = e1y * e2z - e1z * e2y;
        float ny = e1z * e2x - e1x * e2z;
        float nz = e1x * e2y - e1y * e2x;
        float n2 = fmaxf(nx * nx + ny * ny + nz * nz, NORM_EPS_C);
        float inv = 1.0f / sqrtf(n2);
        float* fn = faceNorm + (size_t)idx * 4;
        fn[0] = nx * inv;
        fn[1] = ny * inv;
        fn[2] = nz * inv;
        fn[3] = 0.0f;
    }
}

// ---------------------------------------------------------------------------
// K1: WMMA nearest-face argmin + hinge loss.
//   D = A x B : A(16 faces x K4) packed rows, B(K4 x 16 points),
//   D[m][n] = |c_m|^2 - 2 c_m . p_n   (same argmin as squared distance).
// Face table staged into LDS via the Tensor Data Mover (fallback: manual).
// Double-buffered LDS pipeline, 2 WMMAs/iter, 8 independent argmin chains.
// ---------------------------------------------------------------------------
__global__ void __launch_bounds__(256)
nearest_face_wmma(const float* __restrict__ clothPos,
                  const float* __restrict__ clothPred,
                  const float* __restrict__ packedA,
                  const float4* __restrict__ faceNextC,
                  const float4* __restrict__ faceNorm,
                  float* __restrict__ partials,
                  int B, int Nc, int F, int Fpad, int blocksPerBatch)
{
    extern __shared__ float4 ldsA4[];   // Fpad x 16B packed face rows

    const int b            = blockIdx.x / blocksPerBatch;
    const int blockInBatch = blockIdx.x - b * blocksPerBatch;

    const float4* gA4 = (const float4*)(packedA + (size_t)b * F * 4);
    const int wave = threadIdx.x >> 5;
    const int lane = threadIdx.x & 31;

#if defined(__has_builtin) && __has_builtin(__builtin_amdgcn_tensor_load_to_lds)
    // --- TDM path: one tensor_load_to_lds copies the whole F*16B table ---
    if (wave == 0) {
        unsigned long long gaddr = (unsigned long long)(const void*)gA4;
        unsigned elems = (unsigned)F * 4u;          // 4-byte elements
        unsigned ldsOff = (unsigned)__builtin_amdgcn_groupstaticsize();

        u32x4 g0;
        g0[0] = 1u;                                   // count=1, user descriptor
        g0[1] = ldsOff;                               // lds_addr (bytes)
        g0[2] = (unsigned)(gaddr & 0xFFFFFFFFull);    // global_addr[31:0]
        g0[3] = (unsigned)((gaddr >> 32) & 0x01FFFFFFull) | (2u << 30); // type=2

        i32x8 g1;
        g1[0] = (int)(2u << 16);                      // data_size = 4B
        g1[1] = (int)((elems & 0xFFFFu) << 16);       // tensor_dim0 lo16 @ bit48
        g1[2] = (int)(((elems >> 16) & 0xFFFFu) | (1u << 16)); // dim0 hi | dim1=1
        g1[3] = (int)((elems & 0xFFFFu) << 16);       // tile_dim0 = elems
        g1[4] = 1;                                    // tile_dim1 = 1
        g1[5] = (int)elems;                           // tensor_dim0_stride lo32
        g1[6] = 0;
        g1[7] = 0;

        i32x4 gz = {0, 0, 0, 0};
#if __clang_major__ >= 23
        i32x8 gz8 = {0, 0, 0, 0, 0, 0, 0, 0};
        __builtin_amdgcn_tensor_load_to_lds(g0, g1, gz, gz, gz8, 0);
#else
        __builtin_amdgcn_tensor_load_to_lds(g0, g1, gz, gz, 0);
#endif
        __builtin_amdgcn_s_wait_tensorcnt(0);
    }
    // pad entries (disjoint region; no race with TDM write of [0,F))
    for (int i = F + (int)threadIdx.x; i < Fpad; i += (int)blockDim.x) {
        float4 v; v.x = 0.0f; v.y = 0.0f; v.z = 0.0f; v.w = __builtin_inff();
        ldsA4[i] = v;
    }
#else
    // --- Fallback: cooperative vector staging ---
    for (int i = threadIdx.x; i < Fpad; i += blockDim.x) {
        float4 v;
        if (i < F) v = gA4[i];
        else { v.x = 0.0f; v.y = 0.0f; v.z = 0.0f; v.w = __builtin_inff(); }
        ldsA4[i] = v;
    }
#endif
    __syncthreads();

    const int  laneM = lane & 15;
    const bool hi    = lane >= 16;

    const int  pointBase = blockInBatch * 128 + wave * 16;
    const int  ptRaw     = pointBase + laneM;
    const bool valid     = ptRaw < Nc;
    const int  pt        = valid ? ptRaw : (Nc - 1);

    // B matrix (K4 x N16): lanes 0-15 hold K=0,1 (px,py); lanes 16-31 hold K=2,3 (pz,1)
    const float* pp = clothPos + ((size_t)b * Nc + pt) * 3;
    float px = pp[0], py = pp[1], pz = pp[2];
    v2f bmat;
    bmat[0] = hi ? pz : px;
    bmat[1] = hi ? 1.0f : py;

    // A matrix (M16 x K4): lane L<16 -> {A[L][0],A[L][1]}; lane L>=16 -> {A[L-16][2],A[L-16][3]}
    const v2f* ldsV2   = (const v2f*)ldsA4;
    const int  laneOff = laneM * 2 + (hi ? 1 : 0);   // v2f units within a tile
    const int  mOff    = hi ? 8 : 0;
    const int  tiles   = Fpad >> 4;                  // even (Fpad % 32 == 0)

    // 8 independent argmin chains: per accumulator row keep (min value, tile id)
    float bestV[8];
    int   bestT[8];
#pragma unroll
    for (int r = 0; r < 8; ++r) { bestV[r] = __builtin_inff(); bestT[r] = 0; }

    // Software-pipelined: prefetch tiles t+2/t+3 before the WMMAs of t/t+1
    v2f a0 = ldsV2[laneOff];
    v2f a1 = ldsV2[32 + laneOff];
    int t = 0;
    for (; t + 2 < tiles; t += 2) {
        v2f n0 = ldsV2[(t + 2) * 32 + laneOff];      // ds_loads issue early
        v2f n1 = ldsV2[(t + 3) * 32 + laneOff];

        v8f c0 = {0.f, 0.f, 0.f, 0.f, 0.f, 0.f, 0.f, 0.f};
        v8f d0 = __builtin_amdgcn_wmma_f32_16x16x4_f32(
            false, a0, false, bmat, (short)0, c0, false, false);
        v8f c1 = {0.f, 0.f, 0.f, 0.f, 0.f, 0.f, 0.f, 0.f};
        v8f d1 = __builtin_amdgcn_wmma_f32_16x16x4_f32(
            false, a1, false, bmat, (short)0, c1, false, false);

#pragma unroll
        for (int r = 0; r < 8; ++r) {
            float dv = d0[r];
            if (dv < bestV[r]) { bestV[r] = dv; bestT[r] = t; }
        }
#pragma unroll
        for (int r = 0; r < 8; ++r) {
            float dv = d1[r];
            if (dv < bestV[r]) { bestV[r] = dv; bestT[r] = t + 1; }
        }
        a0 = n0;
        a1 = n1;
    }
    // epilogue: last two tiles
    {
        v8f c0 = {0.f, 0.f, 0.f, 0.f, 0.f, 0.f, 0.f, 0.f};
        v8f d0 = __builtin_amdgcn_wmma_f32_16x16x4_f32(
            false, a0, false, bmat, (short)0, c0, false, false);
        v8f c1 = {0.f, 0.f, 0.f, 0.f, 0.f, 0.f, 0.f, 0.f};
        v8f d1 = __builtin_amdgcn_wmma_f32_16x16x4_f32(
            false, a1, false, bmat, (short)0, c1, false, false);
#pragma unroll
        for (int r = 0; r < 8; ++r) {
            float dv = d0[r];
            if (dv < bestV[r]) { bestV[r] = dv; bestT[r] = t; }
        }
#pragma unroll
        for (int r = 0; r < 8; ++r) {
            float dv = d1[r];
            if (dv < bestV[r]) { bestV[r] = dv; bestT[r] = t + 1; }
        }
    }

    // Resolve face indices, merge the 8 chains (value, then lower index on tie)
    float best    = bestV[0];
    int   bestIdx = bestT[0] * 16 + mOff;
#pragma unroll
    for (int r = 1; r < 8; ++r) {
        int idx = bestT[r] * 16 + mOff + r;
        if (bestV[r] < best || (bestV[r] == best && idx < bestIdx)) {
            best = bestV[r]; bestIdx = idx;
        }
    }

    // Merge the two M-halves (lane p: M=0..7, lane p+16: M=8..15)
    float obest = __shfl_xor(best, 16, 32);
    int   oidx  = __shfl_xor(bestIdx, 16, 32);
    if (obest < best || (obest == best && oidx < bestIdx)) {
        best = obest; bestIdx = oidx;
    }

    // Hinge loss for this point (lanes 0-15 only)
    float contrib = 0.0f;
    if (!hi && valid) {
        const float* np = clothPred + ((size_t)b * Nc + pt) * 3;
        float4 fp = faceNextC[(size_t)b * F + bestIdx];
        float4 fn = faceNorm [(size_t)b * F + bestIdx];
        float dist = (np[0] - fp.x) * fn.x + (np[1] - fp.y) * fn.y + (np[2] - fp.z) * fn.z;
        float tpen = fmaxf(EPS_C - dist, 0.0f);
        contrib = tpen * tpen * tpen;
    }

    // Deterministic wave-level sum
#pragma unroll
    for (int off = 16; off >= 1; off >>= 1)
        contrib += __shfl_xor(contrib, off, 32);

    if (lane == 0)
        partials[blockIdx.x * 8 + wave] = contrib;
}

// ---------------------------------------------------------------------------
// K2: fixed-order reduction of per-wave partials -> scalar loss
// ---------------------------------------------------------------------------
__global__ void __launch_bounds__(256)
final_reduce(const float* __restrict__ partials, float* __restrict__ out,
             int n, float scale)
{
    __shared__ float s[256];
    float acc = 0.0f;
    for (int i = threadIdx.x; i < n; i += 256)
        acc += partials[i];
    s[threadIdx.x] = acc;
    __syncthreads();
#pragma unroll
    for (int off = 128; off > 0; off >>= 1) {
        if (threadIdx.x < off) s[threadIdx.x] += s[threadIdx.x + off];
        __syncthreads();
    }
    if (threadIdx.x == 0)
        out[0] = s[0] * scale;
}

// ---------------------------------------------------------------------------
extern "C" void kernel_launch(void* const* d_in, const int* in_sizes, int n_in,
                              void* d_out, int out_size, void* d_ws, size_t ws_size,
                              hipStream_t stream)
{
    const float* clothPos  = (const float*)d_in[0];   // [B, Nc, 3]
    const float* clothPred = (const float*)d_in[1];   // [B, Nc, 3]
    const float* obsPos    = (const float*)d_in[2];   // [B, No, 3]
    const float* obsNext   = (const float*)d_in[3];   // [B, No, 3]
    const int*   faces     = (const int*)d_in[4];     // [F, 3]

    const int B  = 2;
    const int F  = in_sizes[4] / 3;
    const int Nc = in_sizes[0] / (3 * B);
    const int No = in_sizes[2] / (3 * B);
    const int Fpad = (F + 31) & ~31;      // even number of 16-face tiles

    float* ws        = (float*)d_ws;
    float* packedA   = ws;                                   // B*F*4 floats
    float* faceNextC = packedA   + (size_t)B * F * 4;        // B*F*4 floats
    float* faceNorm  = faceNextC + (size_t)B * F * 4;        // B*F*4 floats
    float* partials  = faceNorm  + (size_t)B * F * 4;        // grid1*8 floats

    {
        int total = B * F;
        int grd = (total + 255) / 256;
        precompute_faces<<<grd, 256, 0, stream>>>(
            obsPos, obsNext, faces, packedA, faceNextC, faceNorm, B, No, F);
    }

    int pointTiles     = (Nc + 15) / 16;
    int blocksPerBatch = (pointTiles + 7) / 8;
    int grid1          = B * blocksPerBatch;
    size_t shmem       = (size_t)Fpad * 16;   // ~160KB for F=10000
    nearest_face_wmma<<<grid1, 256, shmem, stream>>>(
        clothPos, clothPred, packedA,
        (const float4*)faceNextC, (const float4*)faceNorm,
        partials, B, Nc, F, Fpad, blocksPerBatch);

    int nPart = grid1 * 8;
    final_reduce<<<1, 256, 0, stream>>>(partials, (float*)d_out, nPart,
                                        WEIGHT_C / (float)B);
}